// SymbolicGNNLayer_7275674599837
// MI455X (gfx1250) — compile-verified
//
#include <hip/hip_runtime.h>

// ---------------------------------------------------------------------------
// SymbolicGNNLayer on gfx1250 (MI455X): fp32 WMMA (V_WMMA_F32_16X16X4_F32)
// Edge MLP (256->128 relu ->128) + atomic segment-sum + fused GRU node update.
// Edge-feature gather uses double-buffered GLOBAL_LOAD_ASYNC_TO_LDS_B128.
// ---------------------------------------------------------------------------

#define DIM 128
#define FSTR 260   // LDS row stride (floats) for 16x256 feature tiles: bank = (4*row + k) % 64 -> conflict-free
#define HSTR 132   // LDS row stride (floats) for 16x128 tiles

typedef __attribute__((ext_vector_type(2))) float v2f;
typedef __attribute__((ext_vector_type(8))) float v8f;

#if __has_builtin(__builtin_amdgcn_s_wait_asynccnt)
#define ASYNC_WAIT(n) __builtin_amdgcn_s_wait_asynccnt(n)
#else
#define ASYNC_WAIT(n) asm volatile("s_wait_asynccnt %0" ::"i"(n) : "memory")
#endif

// Per-lane async copy of 16 bytes global -> LDS (ASYNCcnt-tracked).
__device__ __forceinline__ void async_g2l_b128(uint32_t lds_byte_off,
                                               const void* gptr) {
    asm volatile("global_load_async_to_lds_b128 %0, %1, off"
                 :
                 : "v"(lds_byte_off), "v"((uint64_t)(uintptr_t)gptr)
                 : "memory");
}

__device__ __forceinline__ v8f wmma_f32_k4(v2f a, v2f b, v8f c) {
    // D = A(16x4) * B(4x16) + C(16x16), all fp32, exact-precision matrix pipe.
    return __builtin_amdgcn_wmma_f32_16x16x4_f32(false, a, false, b, (short)0, c,
                                                 false, false);
}

__device__ __forceinline__ float sigmoidf_fast(float x) {
    return 1.0f / (1.0f + __expf(-x));
}

// ---------------------------------------------------------------------------
// Edge kernel: grid-stride over 16-edge tiles. 8 waves; wave w owns output
// columns [16w, 16w+16). W1 B-fragments live in registers for the whole
// kernel; feature gathers are async double-buffered into LDS.
// ---------------------------------------------------------------------------
__global__ __launch_bounds__(256) void gnn_edge_kernel(
    const float* __restrict__ nf,     // (N,128)
    const int*   __restrict__ edges,  // (E,3) [src, _, tgt]
    const float* __restrict__ W1,     // (128,256) row-major
    const float* __restrict__ b1,     // (128)
    const float* __restrict__ W2,     // (128,128)
    const float* __restrict__ b2,     // (128)
    float*       __restrict__ agg,    // (N,128) pre-zeroed accumulator
    int nEdges, int nTiles)
{
    __shared__ float feat[2][16 * FSTR];  // double-buffered 16x256 concat tiles
    __shared__ float hbuf[16 * HSTR];     // 16x128 hidden tile

    const int tid  = threadIdx.x;
    const int lane = tid & 31;
    const int w    = tid >> 5;      // wave id 0..7 -> N-block
    const int hi   = lane >> 4;     // 0: K-pair {0,1}, 1: K-pair {2,3}
    const int ln   = lane & 15;     // M row (A) / N col (B,C,D)

    // Gather role: thread handles edge-row (tid>>4), 16-float chunk (tid&15).
    const int grow = tid >> 4;
    const int gj   = tid & 15;

    // --- Persistent layer-1 weight fragments (64 K-steps): B(k,n)=W1[(16w+n)*256+k]
    v2f w1f[64];
    {
        const float* bp = W1 + (size_t)(16 * w + ln) * 256 + 2 * hi;
#pragma unroll
        for (int s = 0; s < 64; ++s)
            w1f[s] = *reinterpret_cast<const v2f*>(bp + 4 * s);
    }
    const float bias1 = b1[16 * w + ln];
    const float bias2 = b2[16 * w + ln];
    const float* w2p  = W2 + (size_t)(16 * w + ln) * 128 + 2 * hi;

    // Issue this thread's 4 async b128 copies for one tile into buffer `buf`.
    auto issue_gather = [&](int tileIdx, int buf) {
        int e = tileIdx * 16 + grow;
        if (e >= nEdges) e = nEdges - 1;                  // clamped tail
        const int node = (gj < 8) ? edges[3 * e + 0] : edges[3 * e + 2];
        const float* sp =
            nf + (size_t)node * DIM + ((gj < 8) ? gj * 16 : (gj - 8) * 16);
        const uint32_t dst =
            (uint32_t)(uintptr_t)&feat[buf][grow * FSTR + gj * 16];
#pragma unroll
        for (int q = 0; q < 4; ++q)
            async_g2l_b128(dst + 16u * q, sp + 4 * q);
    };

    int tile = blockIdx.x;
    if (tile < nTiles) issue_gather(tile, 0);            // prologue

    for (int p = 0; tile < nTiles; tile += gridDim.x, p ^= 1) {
        const int ebase    = tile * 16;
        const int nextTile = tile + gridDim.x;

        // ---- overlap: launch next tile's gather, then wait for current one
        if (nextTile < nTiles) {
            issue_gather(nextTile, p ^ 1);
            ASYNC_WAIT(4);   // oldest 4 (current buffer) are complete
        } else {
            ASYNC_WAIT(0);
        }
        __syncthreads();     // current buffer visible to all waves

        // ---- layer 1: h = relu(concat @ W1^T + b1), K = 256 (64 WMMA steps)
        v8f acc = {bias1, bias1, bias1, bias1, bias1, bias1, bias1, bias1};
        {
            const float* ap = &feat[p][ln * FSTR + 2 * hi];
#pragma unroll
            for (int s = 0; s < 64; ++s) {
                v2f a = *reinterpret_cast<const v2f*>(ap + 4 * s);
                acc = wmma_f32_k4(a, w1f[s], acc);
            }
        }
#pragma unroll
        for (int r = 0; r < 8; ++r) acc[r] = fmaxf(acc[r], 0.0f);

        // ---- stage h to LDS (C layout: vgpr r -> row r + 8*hi, col 16w+ln)
#pragma unroll
        for (int r = 0; r < 8; ++r)
            hbuf[(r + 8 * hi) * HSTR + 16 * w + ln] = acc[r];
        __syncthreads();

        // ---- layer 2: msg = h @ W2^T + b2, K = 128 (32 WMMA steps)
        v8f acc2 = {bias2, bias2, bias2, bias2, bias2, bias2, bias2, bias2};
        {
            const float* ap = &hbuf[ln * HSTR + 2 * hi];
#pragma unroll
            for (int s = 0; s < 32; ++s) {
                v2f a = *reinterpret_cast<const v2f*>(ap + 4 * s);
                v2f b = *reinterpret_cast<const v2f*>(w2p + 4 * s);
                acc2 = wmma_f32_k4(a, b, acc2);
            }
        }

        // ---- segment-sum scatter (L2-resident accumulator); tgt from L0-cached
        //      edge records (16 lanes broadcast the same dword).
#pragma unroll
        for (int r = 0; r < 8; ++r) {
            const int row = r + 8 * hi;
            const int e   = ebase + row;
            if (e < nEdges) {
                const int t = edges[3 * e + 2];
                atomicAdd(&agg[(size_t)t * DIM + 16 * w + ln], acc2[r]);
            }
        }
        __syncthreads();     // protects hbuf and the buffer being refilled next
    }
}

// ---------------------------------------------------------------------------
// Node kernel: wave w computes the SAME 16 columns of all six gate chunks so
// the GRU elementwise fuses entirely in registers.
// ---------------------------------------------------------------------------
__global__ __launch_bounds__(256) void gnn_node_kernel(
    const float* __restrict__ nf,    // (N,128)
    const float* __restrict__ agg,   // (N,128)
    const float* __restrict__ Wih,   // (384,128)
    const float* __restrict__ bih,   // (384)
    const float* __restrict__ Whh,   // (384,128)
    const float* __restrict__ bhh,   // (384)
    float*       __restrict__ out,   // (N,128)
    int nNodes, int nTiles)
{
    __shared__ float xls[16 * HSTR];
    __shared__ float als[16 * HSTR];

    const int tid  = threadIdx.x;
    const int lane = tid & 31;
    const int w    = tid >> 5;
    const int hi   = lane >> 4;
    const int ln   = lane & 15;
    const int col  = 16 * w + ln;

    const float bir = bih[col],           bhr = bhh[col];
    const float biz = bih[DIM + col],     bhz = bhh[DIM + col];
    const float bin = bih[2 * DIM + col], bhn = bhh[2 * DIM + col];

    const float* pir = Wih + (size_t)(col) * DIM + 2 * hi;
    const float* piz = Wih + (size_t)(DIM + col) * DIM + 2 * hi;
    const float* pin = Wih + (size_t)(2 * DIM + col) * DIM + 2 * hi;
    const float* phr = Whh + (size_t)(col) * DIM + 2 * hi;
    const float* phz = Whh + (size_t)(DIM + col) * DIM + 2 * hi;
    const float* phn = Whh + (size_t)(2 * DIM + col) * DIM + 2 * hi;

    for (int tile = blockIdx.x; tile < nTiles; tile += gridDim.x) {
        const int nbase = tile * 16;

        // ---- stage x and aggregated tiles (16x128 each), float4 copies
        {
            const int row  = tid >> 4;
            const int j    = tid & 15;              // 8-float chunk
            int node = nbase + row;
            if (node >= nNodes) node = nNodes - 1;  // tail-safe (N%16==0 anyway)
            const float* xs = nf  + (size_t)node * DIM + j * 8;
            const float* as = agg + (size_t)node * DIM + j * 8;
            float* xd = &xls[row * HSTR + j * 8];
            float* ad = &als[row * HSTR + j * 8];
#pragma unroll
            for (int q = 0; q < 2; ++q) {
                *reinterpret_cast<float4*>(xd + 4 * q) =
                    *reinterpret_cast<const float4*>(xs + 4 * q);
                *reinterpret_cast<float4*>(ad + 4 * q) =
                    *reinterpret_cast<const float4*>(as + 4 * q);
            }
        }
        __syncthreads();

        // ---- six GEMM accumulations over K=128 (32 WMMA steps each)
        v8f air = {bir, bir, bir, bir, bir, bir, bir, bir};
        v8f aiz = {biz, biz, biz, biz, biz, biz, biz, biz};
        v8f ain = {bin, bin, bin, bin, bin, bin, bin, bin};
        v8f ahr = {bhr, bhr, bhr, bhr, bhr, bhr, bhr, bhr};
        v8f ahz = {bhz, bhz, bhz, bhz, bhz, bhz, bhz, bhz};
        v8f ahn = {bhn, bhn, bhn, bhn, bhn, bhn, bhn, bhn};
        {
            const float* xap = &xls[ln * HSTR + 2 * hi];
            const float* aap = &als[ln * HSTR + 2 * hi];
#pragma unroll
            for (int s = 0; s < 32; ++s) {
                v2f aa = *reinterpret_cast<const v2f*>(aap + 4 * s);
                v2f ax = *reinterpret_cast<const v2f*>(xap + 4 * s);
                v2f wir = *reinterpret_cast<const v2f*>(pir + 4 * s);
                v2f wiz = *reinterpret_cast<const v2f*>(piz + 4 * s);
                v2f win = *reinterpret_cast<const v2f*>(pin + 4 * s);
                v2f whr = *reinterpret_cast<const v2f*>(phr + 4 * s);
                v2f whz = *reinterpret_cast<const v2f*>(phz + 4 * s);
                v2f whn = *reinterpret_cast<const v2f*>(phn + 4 * s);
                air = wmma_f32_k4(aa, wir, air);
                aiz = wmma_f32_k4(aa, wiz, aiz);
                ain = wmma_f32_k4(aa, win, ain);
                ahr = wmma_f32_k4(ax, whr, ahr);
                ahz = wmma_f32_k4(ax, whz, ahz);
                ahn = wmma_f32_k4(ax, whn, ahn);
            }
        }

        // ---- fused GRU elementwise + store
#pragma unroll
        for (int r = 0; r < 8; ++r) {
            const int row  = r + 8 * hi;
            const int node = nbase + row;
            if (node < nNodes) {
                const float xv = xls[row * HSTR + col];
                const float rg = sigmoidf_fast(air[r] + ahr[r]);
                const float zg = sigmoidf_fast(aiz[r] + ahz[r]);
                const float ng = tanhf(ain[r] + rg * ahn[r]);
                out[(size_t)node * DIM + col] = (1.0f - zg) * ng + zg * xv;
            }
        }
        __syncthreads();
    }
}

// ---------------------------------------------------------------------------
extern "C" void kernel_launch(void* const* d_in, const int* in_sizes, int n_in,
                              void* d_out, int out_size, void* d_ws, size_t ws_size,
                              hipStream_t stream) {
    const float* nf    = (const float*)d_in[0];
    const int*   edges = (const int*)  d_in[1];
    const float* W1    = (const float*)d_in[2];
    const float* b1    = (const float*)d_in[3];
    const float* W2    = (const float*)d_in[4];
    const float* b2    = (const float*)d_in[5];
    const float* Wih   = (const float*)d_in[6];
    const float* bih   = (const float*)d_in[7];
    const float* Whh   = (const float*)d_in[8];
    const float* bhh   = (const float*)d_in[9];
    float* out = (float*)d_out;
    float* agg = (float*)d_ws;                       // (N,128) accumulator

    const int nNodes = in_sizes[0] / DIM;
    const int nEdges = in_sizes[1] / 3;
    const int eTiles = (nEdges + 15) / 16;
    const int nTiles = (nNodes + 15) / 16;

    hipMemsetAsync(agg, 0, (size_t)nNodes * DIM * sizeof(float), stream);

    const int eGrid = eTiles < 2048 ? eTiles : 2048;
    const int nGrid = nTiles < 1024 ? nTiles : 1024;

    gnn_edge_kernel<<<eGrid, 256, 0, stream>>>(nf, edges, W1, b1, W2, b2,
                                               agg, nEdges, eTiles);
    gnn_node_kernel<<<nGrid, 256, 0, stream>>>(nf, agg, Wih, bih, Whh, bhh,
                                               out, nNodes, nTiles);
}